// LinOSS_62302795596555
// MI455X (gfx1250) — compile-verified
//
#include <hip/hip_runtime.h>

// ---------------------------------------------------------------------------
// LinOSS on MI455X (gfx1250, wave32).
//  - GEMMs use V_WMMA_F32_16X16X4_F32 (memory-bound problem: ~26 GFLOP vs
//    ~300 MB traffic -> keep full f32, matrix pipe is nowhere near the limit).
//  - Weight panels staged in LDS (64 KB of the 320 KB WGP pool) with an
//    XOR-16 pair-parity swizzle for conflict-free ds_load_b64 B-fragments.
//  - Each wave computes 16x64 of output (4 tiles): one A-frag load feeds
//    4 WMMAs.
//  - T-scan is a chunked parallel scan (constant 2x2 transition), xs
//    overwrites F in place; F/xs stay L2-resident (192 MB L2).
// ---------------------------------------------------------------------------

#define D_MODEL 256
#define M_OSC   128
#define N_STATE 256            // 2 * M_OSC
#define T_LEN   8192
#define BATCH   8
#define DT      0.01f
#define NCHUNK  64
#define CHUNK_L (T_LEN / NCHUNK)   // 128

#define KDIM    256            // K of both GEMMs
#define COLS_PB 64             // output cols per block (4 N-tiles / wave)
#define ROWS_PB 128            // output rows per block (8 waves * 16)

typedef float v2f __attribute__((ext_vector_type(2)));
typedef float v8f __attribute__((ext_vector_type(8)));

// Stage a 64-col x 256-K panel of a row-major [N][K] weight into LDS.
// LDS layout: 8-byte element {W[col][2p], W[col][2p+1]} at
//   smem[p*64 + (col ^ ((p&1)<<4))]      (XOR-16 swizzle on odd pairs)
// so the K-loop's ds_load_b64 (lanes 0-15: pair kp, lanes 16-31: pair kp+1)
// hits 64 distinct banks.
__device__ __forceinline__ void stage_panel(
    v2f* __restrict__ smem, const float* __restrict__ W, int colBase, int tid)
{
#pragma unroll
  for (int it = 0; it < 16; ++it) {
    int idx = tid + 256 * it;          // 0..4095
    int col = idx >> 6;                // 0..63
    int kq  = idx & 63;                // float4 index within the row
    float4 w = *(const float4*)(W + (size_t)(colBase + col) * KDIM + 4 * kq);
    int p0 = 2 * kq, p1 = 2 * kq + 1;
    smem[p0 * COLS_PB + col]        = v2f{w.x, w.y};   // even pair: no swizzle
    smem[p1 * COLS_PB + (col ^ 16)] = v2f{w.z, w.w};   // odd pair : ^16
  }
}

// ---------------- GEMM 1: F[b,t,n] = sum_d u[b,t,d]*Bw[n,d] + Bb[n] --------
__global__ __launch_bounds__(256) void linoss_gemm_in(
    const float* __restrict__ u, const float* __restrict__ Bw,
    const float* __restrict__ Bb, float* __restrict__ F)
{
  __shared__ v2f smem[(KDIM / 2) * COLS_PB];   // 64 KB

  const int tid  = threadIdx.x;
  const int lane = tid & 31;
  const int wave = tid >> 5;
  const int rowBase = blockIdx.x * ROWS_PB + wave * 16;
  const int colBase = blockIdx.y * COLS_PB;
  const int r  = lane & 15;
  const int kh = lane >> 4;

  stage_panel(smem, Bw, colBase, tid);
  __syncthreads();

  const float* arow = u + (size_t)(rowBase + r) * KDIM + 2 * kh;
  v8f acc0 = {}, acc1 = {}, acc2 = {}, acc3 = {};

#pragma unroll 4
  for (int k0 = 0; k0 < KDIM; k0 += 4) {
    const int kp  = k0 >> 1;                 // even pair base; lane pair=kp+kh
    const v2f* brow = smem + (size_t)(kp + kh) * COLS_PB;
    const int  csw  = r ^ (kh << 4);         // pair parity == kh
    v2f a  = *(const v2f*)(arow + k0);       // one A-frag feeds 4 WMMAs
    v2f b0 = brow[csw];
    v2f b1 = brow[16 ^ csw];
    v2f b2 = brow[32 ^ csw];
    v2f b3 = brow[48 ^ csw];
    acc0 = __builtin_amdgcn_wmma_f32_16x16x4_f32(false, a, false, b0, (short)0, acc0, false, false);
    acc1 = __builtin_amdgcn_wmma_f32_16x16x4_f32(false, a, false, b1, (short)0, acc1, false, false);
    acc2 = __builtin_amdgcn_wmma_f32_16x16x4_f32(false, a, false, b2, (short)0, acc2, false, false);
    acc3 = __builtin_amdgcn_wmma_f32_16x16x4_f32(false, a, false, b3, (short)0, acc3, false, false);
  }

  // C/D layout: acc[i] -> (m = i + 8*kh, n = r) within each 16x16 tile
  v8f accs[4] = {acc0, acc1, acc2, acc3};
#pragma unroll
  for (int nt = 0; nt < 4; ++nt) {
    const int   col  = colBase + nt * 16 + r;
    const float bias = Bb[col];
    float* op = F + (size_t)(rowBase + 8 * kh) * N_STATE + col;
#pragma unroll
    for (int i = 0; i < 8; ++i)
      op[(size_t)i * N_STATE] = accs[nt][i] + bias;
  }
}

// ---------------- scan coefficients per oscillator j -----------------------
__device__ __forceinline__ void linoss_coeffs(
    const float* __restrict__ lw, const float* __restrict__ zl, int j,
    float& czz, float& czy, float& cyz, float& cyy)
{
  float omega = fminf(fmaxf(expf(lw[j]), 1.0e-4f), 100.0f);
  float A = omega * omega;
  float S = 1.0f / (1.0f + DT * DT * A);
  float d = 1.0f / (1.0f + expf(zl[j]));       // 1 - sigmoid(zl)
  czz = d * S;
  czy = -d * DT * A * S;
  cyz = d * DT * S;
  cyy = d * S;
}

// ---------------- Phase 1: per-chunk end states (zero init) ----------------
__global__ __launch_bounds__(M_OSC) void linoss_scan_ends(
    const float* __restrict__ F, const float* __restrict__ lw,
    const float* __restrict__ zl, float* __restrict__ E)
{
  const int j = threadIdx.x, c = blockIdx.x, b = blockIdx.y;
  float czz, czy, cyz, cyy;
  linoss_coeffs(lw, zl, j, czz, czy, cyz, cyy);

  float z = 0.f, y = 0.f;
  const float* base = F + ((size_t)b * T_LEN + (size_t)c * CHUNK_L) * N_STATE;
  for (int i = 0; i < CHUNK_L; ++i) {
    float fz = base[(size_t)i * N_STATE + j];
    float fy = base[(size_t)i * N_STATE + M_OSC + j];
    float zn = fmaf(czz, z, fmaf(czy, y, fz));
    float yn = fmaf(cyz, z, fmaf(cyy, y, fy));
    z = zn; y = yn;
  }
  float* e = E + ((size_t)b * NCHUNK + c) * N_STATE;
  e[j] = z; e[M_OSC + j] = y;
}

// ---------------- Phase 2: sequential carry over chunks --------------------
__global__ __launch_bounds__(M_OSC) void linoss_scan_carry(
    const float* __restrict__ E, const float* __restrict__ lw,
    const float* __restrict__ zl, float* __restrict__ carry)
{
  const int j = threadIdx.x, b = blockIdx.x;
  float czz, czy, cyz, cyy;
  linoss_coeffs(lw, zl, j, czz, czy, cyz, cyy);

  // P = M^CHUNK_L via 128 register-only 2x2 multiplies
  float p00 = 1.f, p01 = 0.f, p10 = 0.f, p11 = 1.f;
  for (int i = 0; i < CHUNK_L; ++i) {
    float q00 = fmaf(czz, p00, czy * p10);
    float q01 = fmaf(czz, p01, czy * p11);
    float q10 = fmaf(cyz, p00, cyy * p10);
    float q11 = fmaf(cyz, p01, cyy * p11);
    p00 = q00; p01 = q01; p10 = q10; p11 = q11;
  }

  float z = 0.f, y = 0.f;                       // carry entering chunk 0
  for (int c = 0; c < NCHUNK; ++c) {
    float* cp = carry + ((size_t)b * NCHUNK + c) * N_STATE;
    cp[j] = z; cp[M_OSC + j] = y;
    const float* e = E + ((size_t)b * NCHUNK + c) * N_STATE;
    float zn = fmaf(p00, z, fmaf(p01, y, e[j]));
    float yn = fmaf(p10, z, fmaf(p11, y, e[M_OSC + j]));
    z = zn; y = yn;
  }
}

// ---------------- Phase 3: full scan per chunk, xs overwrites F ------------
__global__ __launch_bounds__(M_OSC) void linoss_scan_apply(
    float* __restrict__ F, const float* __restrict__ lw,
    const float* __restrict__ zl, const float* __restrict__ carry)
{
  const int j = threadIdx.x, c = blockIdx.x, b = blockIdx.y;
  float czz, czy, cyz, cyy;
  linoss_coeffs(lw, zl, j, czz, czy, cyz, cyy);

  const float* cp = carry + ((size_t)b * NCHUNK + c) * N_STATE;
  float z = cp[j], y = cp[M_OSC + j];

  float* base = F + ((size_t)b * T_LEN + (size_t)c * CHUNK_L) * N_STATE;
  for (int i = 0; i < CHUNK_L; ++i) {
    float fz = base[(size_t)i * N_STATE + j];
    float fy = base[(size_t)i * N_STATE + M_OSC + j];
    float zn = fmaf(czz, z, fmaf(czy, y, fz));
    float yn = fmaf(cyz, z, fmaf(cyy, y, fy));
    z = zn; y = yn;
    base[(size_t)i * N_STATE + j]         = z;   // F[t] consumed this step:
    base[(size_t)i * N_STATE + M_OSC + j] = y;   // in-place is safe
  }
}

// ---------------- GEMM 2: y = xs @ C^T + u @ D^T ---------------------------
__global__ __launch_bounds__(256) void linoss_gemm_out(
    const float* __restrict__ xs, const float* __restrict__ u,
    const float* __restrict__ C, const float* __restrict__ Dm,
    float* __restrict__ out)
{
  __shared__ v2f smem[(KDIM / 2) * COLS_PB];   // 64 KB, reused for C then D

  const int tid  = threadIdx.x;
  const int lane = tid & 31;
  const int wave = tid >> 5;
  const int rowBase = blockIdx.x * ROWS_PB + wave * 16;
  const int colBase = blockIdx.y * COLS_PB;
  const int r  = lane & 15;
  const int kh = lane >> 4;
  const int csw = r ^ (kh << 4);

  v8f acc0 = {}, acc1 = {}, acc2 = {}, acc3 = {};

  // ---- y += xs @ C^T ----
  stage_panel(smem, C, colBase, tid);
  __syncthreads();
  {
    const float* arow = xs + (size_t)(rowBase + r) * N_STATE + 2 * kh;
#pragma unroll 4
    for (int k0 = 0; k0 < KDIM; k0 += 4) {
      const v2f* brow = smem + (size_t)((k0 >> 1) + kh) * COLS_PB;
      v2f a  = *(const v2f*)(arow + k0);
      v2f b0 = brow[csw];
      v2f b1 = brow[16 ^ csw];
      v2f b2 = brow[32 ^ csw];
      v2f b3 = brow[48 ^ csw];
      acc0 = __builtin_amdgcn_wmma_f32_16x16x4_f32(false, a, false, b0, (short)0, acc0, false, false);
      acc1 = __builtin_amdgcn_wmma_f32_16x16x4_f32(false, a, false, b1, (short)0, acc1, false, false);
      acc2 = __builtin_amdgcn_wmma_f32_16x16x4_f32(false, a, false, b2, (short)0, acc2, false, false);
      acc3 = __builtin_amdgcn_wmma_f32_16x16x4_f32(false, a, false, b3, (short)0, acc3, false, false);
    }
  }

  // ---- y += u @ D^T ----
  __syncthreads();                       // all waves done reading C panel
  stage_panel(smem, Dm, colBase, tid);
  __syncthreads();
  {
    const float* arow = u + (size_t)(rowBase + r) * D_MODEL + 2 * kh;
#pragma unroll 4
    for (int k0 = 0; k0 < KDIM; k0 += 4) {
      const v2f* brow = smem + (size_t)((k0 >> 1) + kh) * COLS_PB;
      v2f a  = *(const v2f*)(arow + k0);
      v2f b0 = brow[csw];
      v2f b1 = brow[16 ^ csw];
      v2f b2 = brow[32 ^ csw];
      v2f b3 = brow[48 ^ csw];
      acc0 = __builtin_amdgcn_wmma_f32_16x16x4_f32(false, a, false, b0, (short)0, acc0, false, false);
      acc1 = __builtin_amdgcn_wmma_f32_16x16x4_f32(false, a, false, b1, (short)0, acc1, false, false);
      acc2 = __builtin_amdgcn_wmma_f32_16x16x4_f32(false, a, false, b2, (short)0, acc2, false, false);
      acc3 = __builtin_amdgcn_wmma_f32_16x16x4_f32(false, a, false, b3, (short)0, acc3, false, false);
    }
  }

  v8f accs[4] = {acc0, acc1, acc2, acc3};
#pragma unroll
  for (int nt = 0; nt < 4; ++nt) {
    float* op = out + (size_t)(rowBase + 8 * kh) * D_MODEL + colBase + nt * 16 + r;
#pragma unroll
    for (int i = 0; i < 8; ++i)
      __builtin_nontemporal_store(accs[nt][i], op + (size_t)i * D_MODEL);
  }
}

// ---------------------------------------------------------------------------
extern "C" void kernel_launch(void* const* d_in, const int* in_sizes, int n_in,
                              void* d_out, int out_size, void* d_ws, size_t ws_size,
                              hipStream_t stream)
{
  const float* u  = (const float*)d_in[0];   // (8, 8192, 256)
  const float* lw = (const float*)d_in[1];   // (128,)
  const float* zl = (const float*)d_in[2];   // (128,)
  const float* Bw = (const float*)d_in[3];   // (256, 256)
  const float* Bb = (const float*)d_in[4];   // (256,)
  const float* C  = (const float*)d_in[5];   // (256, 256)
  const float* Dm = (const float*)d_in[6];   // (256, 256)
  float* out = (float*)d_out;

  // Workspace layout: F/xs (64 MB) | chunk end-states | chunk carries
  float* F     = (float*)d_ws;
  float* E     = F + (size_t)BATCH * T_LEN * N_STATE;
  float* carry = E + (size_t)BATCH * NCHUNK * N_STATE;

  const dim3 g1(BATCH * T_LEN / ROWS_PB, N_STATE / COLS_PB);   // (512, 4)
  linoss_gemm_in<<<g1, 256, 0, stream>>>(u, Bw, Bb, F);

  linoss_scan_ends <<<dim3(NCHUNK, BATCH), M_OSC, 0, stream>>>(F, lw, zl, E);
  linoss_scan_carry<<<BATCH,               M_OSC, 0, stream>>>(E, lw, zl, carry);
  linoss_scan_apply<<<dim3(NCHUNK, BATCH), M_OSC, 0, stream>>>(F, lw, zl, carry);

  const dim3 g2(BATCH * T_LEN / ROWS_PB, D_MODEL / COLS_PB);   // (512, 4)
  linoss_gemm_out<<<g2, 256, 0, stream>>>(F, u, C, Dm, out);
}